// QGate_65481071401635
// MI455X (gfx1250) — compile-verified
//
#include <hip/hip_runtime.h>
#include <math.h>

// Analytic collapse of the reference:
//   out[b,q] = prod_{i=0}^{NQ-1-q} cos(x[b,i])
// (product state after RX; CNOT ladder => prefix-parity permutation;
//  <Z_q> telescopes to a prefix product of cos; output index reversed by
//  the Pauli label convention.)
//
// Memory: 2048x14 f32 in, 2048x14 f32 out (229 KB total) -> pure latency
// bound on MI455X (23.3 TB/s). No matmul structure survives, so no WMMA;
// instead we use the CDNA5 async-copy path (GLOBAL_LOAD_ASYNC_TO_LDS_B128 +
// s_wait_asynccnt) to stage each block's tile with fully coalesced 128-bit
// lane transfers, then read conflict-free rows out of LDS.

#define NQ 14
#define ROWS_PER_BLOCK 128   // 4 waves of 32; tile = 128*14*4 = 7168 B = 448 x 16B chunks

__global__ __launch_bounds__(ROWS_PER_BLOCK)
void qcirc_expz_kernel(const float* __restrict__ x,
                       float* __restrict__ out,
                       int batch)
{
    __shared__ float tile[ROWS_PER_BLOCK * NQ];

    const int tid  = (int)threadIdx.x;
    const int row0 = (int)blockIdx.x * ROWS_PER_BLOCK;
    int rows = batch - row0;
    if (rows > ROWS_PER_BLOCK) rows = ROWS_PER_BLOCK;

    // ---- Stage x tile into LDS with CDNA5 async copies (ASYNCcnt path) ----
    // GVS addressing: saddr = 64-bit SGPR base of this block's tile,
    // vaddr = per-lane byte offset, vdst = per-lane LDS byte address.
    const unsigned tileBytes = (rows > 0) ? (unsigned)rows * (NQ * 4u) : 0u;
    const unsigned lds_base  = (unsigned)(uintptr_t)&tile[0];
    const unsigned long long gbase =
        (unsigned long long)(uintptr_t)(x + (size_t)row0 * NQ);

    #pragma unroll
    for (int j = 0; j < (ROWS_PER_BLOCK * NQ * 4 / 16 + ROWS_PER_BLOCK - 1) / ROWS_PER_BLOCK; ++j) {
        unsigned off = (unsigned)(tid + j * ROWS_PER_BLOCK) * 16u;
        if (off < tileBytes) {
            unsigned dst = lds_base + off;
            asm volatile("global_load_async_to_lds_b128 %0, %1, %2"
                         :: "v"(dst), "v"(off), "s"(gbase)
                         : "memory");
        }
    }

    // Each wave drains its own ASYNCcnt, then the workgroup barrier makes
    // the whole tile visible to every wave.
#if __has_builtin(__builtin_amdgcn_s_wait_asynccnt)
    __builtin_amdgcn_s_wait_asynccnt(0);
#else
    asm volatile("s_wait_asynccnt 0x0" ::: "memory");
#endif
    __syncthreads();

    // ---- Per-row reversed cumulative product of cosines ----
    // LDS row stride = 14 floats = 56 B; across a wave32 the row bases hit
    // all 32 even banks exactly once -> conflict-free.
    if (tid < rows) {
        const float* r = &tile[tid * NQ];
        float pref[NQ];
        float p = 1.0f;
        #pragma unroll
        for (int i = 0; i < NQ; ++i) {
            p *= cosf(r[i]);     // accurate ocml cos: 14-term product amplifies error
            pref[i] = p;
        }
        float* op = out + (size_t)(row0 + tid) * NQ;
        #pragma unroll
        for (int q = 0; q < NQ; ++q) {
            op[q] = pref[NQ - 1 - q];   // out[b,q] = prefix(NQ-1-q)
        }
    }
}

extern "C" void kernel_launch(void* const* d_in, const int* in_sizes, int n_in,
                              void* d_out, int out_size, void* d_ws, size_t ws_size,
                              hipStream_t stream)
{
    const float* x  = (const float*)d_in[0];
    float* out      = (float*)d_out;
    const int total = in_sizes[0];          // BATCH * NQ
    const int batch = total / NQ;           // 2048
    const int grid  = (batch + ROWS_PER_BLOCK - 1) / ROWS_PER_BLOCK;  // 16 blocks
    qcirc_expz_kernel<<<grid, ROWS_PER_BLOCK, 0, stream>>>(x, out, batch);
}